// Grouping_34273839022361
// MI455X (gfx1250) — compile-verified
//
#include <hip/hip_runtime.h>

typedef __attribute__((ext_vector_type(2))) float v2f;
typedef __attribute__((ext_vector_type(4))) float v4f;
typedef __attribute__((ext_vector_type(8))) float v8f;
typedef unsigned long long u64;

#define Bn 8
#define Nn 8192
#define Sn 2048
#define Dn 64
#define Kn 32
#define TQ 16          // queries per tile (WMMA M)
#define CHUNK 256      // columns (points) per LDS chunk
#define NWAVES 16      // waves per block = CHUNK/16 col-tiles
#define NCHUNK (Nn / CHUNK)

// Monotonic float -> u32 map (total order, ascending)
__device__ __forceinline__ unsigned orderedBits(float f) {
    unsigned u = __float_as_uint(f);
    return u ^ ((unsigned)((int)u >> 31) | 0x80000000u);
}

// Wave-wide max over u64 keys; result uniform across all 32 lanes.
__device__ __forceinline__ u64 waveMaxU64(u64 k) {
#pragma unroll
    for (int off = 16; off > 0; off >>= 1) {
        u64 o = __shfl_xor(k, off);
        if (o > k) k = o;
    }
    return k;
}

__global__ __launch_bounds__(512)
void knn_group_kernel(const float* __restrict__ xyz,      // [B,N,3]
                      const float* __restrict__ f,        // [B,N,D]
                      const float* __restrict__ xyzs,     // [B,S,3]
                      float* __restrict__ out)            // xyz_grouped ++ f_grouped
{
    __shared__ float scores[2 * TQ * CHUNK];   // 32 KB, double buffered

    const int lane = threadIdx.x & 31;
    const int wave = threadIdx.x >> 5;
    const int b    = blockIdx.x / (Sn / TQ);
    const int q0   = (blockIdx.x % (Sn / TQ)) * TQ;

    // ---- A operand: 16 queries x 4 (qx,qy,qz,1) ----
    // 32-bit A 16x4 layout: lanes 0-15 hold M=lane, VGPR0=K0, VGPR1=K1;
    //                       lanes 16-31 hold M=lane-16, VGPR0=K2, VGPR1=K3.
    v2f A;
    {
        const float* q = xyzs + ((size_t)b * Sn + q0 + (lane & 15)) * 3;
        float qx = q[0], qy = q[1], qz = q[2];
        A[0] = (lane < 16) ? qx : qz;
        A[1] = (lane < 16) ? qy : 1.0f;
    }

    // ---- per-wave top-32: lane l owns one (val,idx) packed slot key ----
    u64 slotKey = 0xFF800000FFFFFFFFull;    // (+INF mapped)<<32 | ~0
    u64 thrKey  = waveMaxU64(slotKey);      // current worst of the 32 (uniform)

    const float* pbase = xyz + (size_t)b * Nn * 3;

    // ---- prologue: load chunk 0's points ----
    float px, py, pz;
    {
        const float* p = pbase + (size_t)(wave * 16 + (lane & 15)) * 3;
        px = p[0]; py = p[1]; pz = p[2];
    }

    for (int it = 0; it < NCHUNK; ++it) {
        const int c0 = it * CHUNK;
        // ---- B operand (-2px,-2py,-2pz,|p|^2), WMMA, stage D into LDS ----
        {
            v2f Bv;
            Bv[0] = (lane < 16) ? (-2.0f * px) : (-2.0f * pz);
            Bv[1] = (lane < 16) ? (-2.0f * py) : (px * px + py * py + pz * pz);
            v8f Cz = {};
            // D[m][n] = -2*q.p + |p|^2  (rank-equivalent to squared distance)
            v8f Dm = __builtin_amdgcn_wmma_f32_16x16x4_f32(
                false, A, false, Bv, (short)0, Cz, false, false);
            float* sb = &scores[(it & 1) * (TQ * CHUNK)];
            int col   = wave * 16 + (lane & 15);
            int mbase = (lane >> 4) * 8;
#pragma unroll
            for (int r = 0; r < 8; ++r)
                sb[(mbase + r) * CHUNK + col] = Dm[r];
        }
        // ---- software pipeline: issue next chunk's point load now; the
        //      selection phase below hides its latency ----
        if (it + 1 < NCHUNK) {
            const float* p =
                pbase + (size_t)((it + 1) * CHUNK + wave * 16 + (lane & 15)) * 3;
            px = p[0]; py = p[1]; pz = p[2];
        }
        __syncthreads();

        // ---- selection: wave w streams row w of the chunk, 4 cands/lane/load ----
        const float* srow = &scores[(it & 1) * (TQ * CHUNK) + wave * CHUNK];
#pragma unroll 1
        for (int j = 0; j < CHUNK / 128; ++j) {
            int cb = j * 128 + 4 * lane;
            v4f s4 = *(const v4f*)(srow + cb);          // ds_load_b128
            u64 k[4];
#pragma unroll
            for (int t = 0; t < 4; ++t)
                k[t] = ((u64)orderedBits(s4[t]) << 32) | (unsigned)(c0 + cb + t);
            // fast reject: one ballot for the whole 128-candidate group
            bool any = (k[0] < thrKey) | (k[1] < thrKey) |
                       (k[2] < thrKey) | (k[3] < thrKey);
            if (__ballot(any) == 0ull) continue;
#pragma unroll
            for (int t = 0; t < 4; ++t) {
                u64 candKey = k[t];
                for (;;) {
                    unsigned long long m = __ballot(candKey < thrKey);
                    if (m == 0ull) break;
                    int src = __ffsll((long long)m) - 1;
                    u64 ck = __shfl(candKey, src);
                    // evict current worst slot (lowest lane holding thrKey)
                    unsigned long long rm = __ballot(slotKey == thrKey);
                    int rs = __ffsll((long long)rm) - 1;
                    if (lane == rs)  slotKey = ck;
                    if (lane == src) candKey = ~0ull;   // consumed
                    thrKey = waveMaxU64(slotKey);
                }
            }
        }
        // no tail barrier: score staging is double-buffered
    }

    // ---- bitonic sort across lanes, ascending key = (val, idx) -> top_k order ----
    u64 key = slotKey;
#pragma unroll
    for (int k = 2; k <= 32; k <<= 1) {
#pragma unroll
        for (int j = k >> 1; j > 0; j >>= 1) {
            u64 ok = __shfl_xor(key, j);
            bool takeMin = (((lane & k) == 0) == ((lane & j) == 0));
            if ((ok < key) == takeMin) key = ok;
        }
    }
    unsigned idx = (unsigned)key;   // neighbor index, rank = lane

    // ---- gather + write ----
    const int q = q0 + wave;
    {   // xyz_grouped[b][q][lane][0..2]
        float* oxyz = out + (((size_t)b * Sn + q) * Kn + lane) * 3;
        const float* sp = pbase + (size_t)idx * 3;
        oxyz[0] = sp[0]; oxyz[1] = sp[1]; oxyz[2] = sp[2];
    }
    const size_t F_OFF = (size_t)Bn * Sn * Kn * 3;
    const float* fb = f + (size_t)b * Nn * Dn;
    float* of = out + F_OFF + ((size_t)b * Sn + q) * (size_t)(Kn * Dn);
#pragma unroll 4
    for (int k = 0; k < Kn; ++k) {
        unsigned id = __shfl(idx, k);
        const float* srcrow = fb + (size_t)id * Dn;
        float*       dstrow = of + (size_t)k * Dn;
        dstrow[lane]      = srcrow[lane];    // two fully coalesced 128B lines
        dstrow[lane + 32] = srcrow[lane + 32];
    }
}

extern "C" void kernel_launch(void* const* d_in, const int* in_sizes, int n_in,
                              void* d_out, int out_size, void* d_ws, size_t ws_size,
                              hipStream_t stream) {
    (void)in_sizes; (void)n_in; (void)d_ws; (void)ws_size; (void)out_size;
    const float* xyz  = (const float*)d_in[0];
    const float* f    = (const float*)d_in[1];
    const float* xyzs = (const float*)d_in[2];
    // d_in[3] (f_sampled) is unused by the reference outputs.
    dim3 grid(Bn * (Sn / TQ));   // 1024 blocks
    dim3 block(NWAVES * 32);     // 512 threads = 16 wave32
    knn_group_kernel<<<grid, block, 0, stream>>>(xyz, f, xyzs, (float*)d_out);
}